// Model_34772055229095
// MI455X (gfx1250) — compile-verified
//
#include <hip/hip_runtime.h>
#include <hip/hip_bf16.h>
#include <math.h>

// ---------------------------------------------------------------------------
// S-Mamba forward, fully fused: one workgroup (8 wave32) per batch element.
// All activations live in LDS (134 KB / WGP of the 320 KB CDNA5 pool).
// All GEMMs use V_WMMA_F32_16X16X4_F32 (fp32 matrix core, exact reference
// numerics). Weights (~4.6 MB total) stay hot in the 192 MB L2; HBM traffic
// is 3.7 MB in + 8 KB out -> latency-bound, single launch.
//
// Address spaces are explicit: weights -> addrspace(1) (global_load_b64),
// activations -> addrspace(3) (ds_load_b64 with immediate offsets, 32-bit
// address math). WMMA inner loop is branch-free: A/W row overreads land only
// in store-masked rows/cols; only the K-tail is zero-masked via cndmask.
// ---------------------------------------------------------------------------

typedef __attribute__((ext_vector_type(2))) float v2f;
typedef __attribute__((ext_vector_type(8))) float v8f;

typedef __attribute__((address_space(1))) const float* gfp;
typedef __attribute__((address_space(1))) const v2f*   gv2fp;
typedef __attribute__((address_space(3))) float*       lfp;
typedef __attribute__((address_space(3))) const v2f*   lv2fp;

__device__ __forceinline__ gfp as_global(const float* p) {
  return (gfp)(unsigned long long)p;          // inttoptr -> addrspace(1)
}
__device__ __forceinline__ lfp as_lds(float* p) {
  // generic->LDS is truncation to addr[31:0] (ISA 10.2); pointers never null
  return (lfp)(unsigned int)(unsigned long long)p;
}

#define B_SZ     2048
#define SEQ_LEN  32
#define ENC_IN   14
#define D_MODEL  256
#define SP_D     14
#define D_STATE  8
#define D_FF     256
#define DT_T     16
#define DT_S     1
#define NWAVE    8
#define NTHREADS 256

// LDS layout (floats)
#define OFF_X    0
#define OFF_XZ   (OFF_X    + 3584)   // L*2D  (14*512 == 256*28 == 7168)
#define OFF_XI   (OFF_XZ   + 7168)
#define OFF_DT   (OFF_XI   + 3584)
#define OFF_MO   (OFF_DT   + 3584)
#define OFF_XF   (OFF_MO   + 3584)
#define OFF_XDBL (OFF_XF   + 3584)   // max(14*32, 256*17) = 4352
#define OFF_H    (OFF_XDBL + 4352)   // 16 x 256 FFN hidden tile
#define OFF_RED  (OFF_H    + 4096)
#define LDS_FLOATS (OFF_RED + 32)    // 33568 floats = 134272 bytes

struct LayerW {
  const float *c1b, *c1w, *c2b, *c2w;
  const float *ln1b, *ln1g, *ln2b, *ln2g;
  // mb (backward mamba)
  const float *Alog_b, *D_b, *convb_b, *convw_b, *dtb_b, *dtw_b,
              *inp_b, *outp_b, *xproj_b;
  // mf (forward mamba)
  const float *Alog_f, *D_f, *convb_f, *convw_f, *dtb_f, *dtw_f,
              *inp_f, *outp_f, *xproj_f;
};

struct ModelP {
  const float *emb_b, *emb_w, *fin_b, *fin_w, *pos;
  LayerW sl[2];
  const float *space_nb, *space_ng;
  LayerW tl[2];
  const float *time_nb, *time_ng;
  const float *x_enc;
  float *out;
};

// ---------------------------------------------------------------------------

__device__ __forceinline__ float siluf(float x) { return x / (1.f + __expf(-x)); }
__device__ __forceinline__ float geluf(float x) {
  const float c = 0.7978845608028654f; // sqrt(2/pi), tanh-approx gelu (jax default)
  float t = tanhf(c * (x + 0.044715f * x * x * x));
  return 0.5f * x * (1.f + t);
}
__device__ __forceinline__ float softplusf(float x) {
  return (x > 20.f) ? x : log1pf(__expf(x));
}

// One 16x16 f32 tile: acc += A[16 x K] * W[16 rows, K]^T.
// A (LDS, addrspace(3)): unguarded row reads -- junk rows pollute only
// store-masked outputs. W (global): row clamped to 0 beyond Wr (stays inside
// the allocation). K-tail (K % 4): zero-masked via clamped index + cndmask.
template <int K>
__device__ __forceinline__ v8f wmma16(lfp A, int lda,
                                      const float* W, int ldw, int Wr,
                                      v8f acc)
{
  const int lane = threadIdx.x & 31;
  const int r  = lane & 15;
  const int hh = lane >> 4;
  lfp Arow = A + r * lda;
  gfp Wrow = as_global(W) + ((r < Wr) ? r : 0) * ldw;
  constexpr int KM = K & ~3;
#pragma unroll 8
  for (int k0 = 0; k0 < KM; k0 += 4) {
    const int ka = k0 + 2 * hh;
    v2f a = *(lv2fp)(Arow + ka);
    v2f b = *(gv2fp)(Wrow + ka);
    acc = __builtin_amdgcn_wmma_f32_16x16x4_f32(false, a, false, b,
                                                (short)0, acc, false, false);
  }
  if (K & 3) {
    const int ka  = KM + 2 * hh;
    const int k0c = (ka     < K) ? ka     : 0;
    const int k1c = (ka + 1 < K) ? ka + 1 : 0;
    const float a0 = Arow[k0c], a1 = Arow[k1c];
    const float b0 = Wrow[k0c], b1 = Wrow[k1c];
    v2f a = { (ka < K) ? a0 : 0.f, (ka + 1 < K) ? a1 : 0.f };
    v2f b = { (ka < K) ? b0 : 0.f, (ka + 1 < K) ? b1 : 0.f };
    acc = __builtin_amdgcn_wmma_f32_16x16x4_f32(false, a, false, b,
                                                (short)0, acc, false, false);
  }
  return acc;
}

// Y[MxN] (LDS) = act(A[MxK] (LDS) @ W[NxK]^T (global) + bias); tiles / waves.
// ACT: 0 none, 1 silu, 2 gelu, 3 softplus. Dims compile-time -> guards fold.
template <int ACT, int M, int K, int N>
__device__ void gemm_lds(lfp A, int lda,
                         const float* W, const float* bias,
                         lfp Y, int ldy)
{
  const int wid  = threadIdx.x >> 5;
  const int lane = threadIdx.x & 31;
  const int r = lane & 15, hh = lane >> 4;
  constexpr int mt = (M + 15) >> 4, nt = (N + 15) >> 4;
  for (int t = wid; t < mt * nt; t += NWAVE) {
    const int im = t / nt, in = t % nt;
    v8f acc = {};
    acc = wmma16<K>(A + im * 16 * lda, lda,
                    W + in * 16 * K, K, N - in * 16, acc);
    const int n = in * 16 + r;
    if (n < N) {
      const float bv = bias ? as_global(bias)[n] : 0.f;
#pragma unroll
      for (int v = 0; v < 8; ++v) {
        const int m = im * 16 + v + 8 * hh;
        if (m < M) {
          float x = acc[v] + bv;
          if (ACT == 1) x = siluf(x);
          else if (ACT == 2) x = geluf(x);
          else if (ACT == 3) x = softplusf(x);
          Y[m * ldy + n] = x;
        }
      }
    }
  }
}

// Y[rev(m)][n] += A @ W^T (out_proj accumulate; rev un-flips the bwd mamba)
template <int M, int K, int N>
__device__ void gemm_acc_rev(lfp A, int lda, const float* W,
                             lfp Y, int ldy, bool rev)
{
  const int wid  = threadIdx.x >> 5;
  const int lane = threadIdx.x & 31;
  const int r = lane & 15, hh = lane >> 4;
  constexpr int mt = (M + 15) >> 4, nt = (N + 15) >> 4;
  for (int t = wid; t < mt * nt; t += NWAVE) {
    const int im = t / nt, in = t % nt;
    v8f acc = {};
    acc = wmma16<K>(A + im * 16 * lda, lda,
                    W + in * 16 * K, K, N - in * 16, acc);
    const int n = in * 16 + r;
    if (n < N) {
#pragma unroll
      for (int v = 0; v < 8; ++v) {
        const int m = im * 16 + v + 8 * hh;
        if (m < M) {
          const int mr = rev ? (M - 1 - m) : m;
          Y[mr * ldy + n] += acc[v];
        }
      }
    }
  }
}

// In-place LayerNorm over rows of length D (wave-per-row for wide D).
template <int L, int D>
__device__ void layernorm_rows(lfp x, const float* g, const float* b)
{
  const int tid = threadIdx.x;
  gfp gg = as_global(g), bg = as_global(b);
  if (D >= 32) {
    const int wid = tid >> 5, lane = tid & 31;
    for (int row = wid; row < L; row += NWAVE) {
      lfp xr = x + row * D;
      float s = 0.f;
      for (int i = lane; i < D; i += 32) s += xr[i];
#pragma unroll
      for (int o = 16; o > 0; o >>= 1) s += __shfl_xor(s, o, 32);
      const float mean = s / (float)D;
      float v = 0.f;
      for (int i = lane; i < D; i += 32) { float d0 = xr[i] - mean; v += d0 * d0; }
#pragma unroll
      for (int o = 16; o > 0; o >>= 1) v += __shfl_xor(v, o, 32);
      const float rstd = rsqrtf(v / (float)D + 1e-5f);
      for (int i = lane; i < D; i += 32)
        xr[i] = (xr[i] - mean) * rstd * gg[i] + bg[i];
    }
  } else {
    for (int row = tid; row < L; row += NTHREADS) {
      lfp xr = x + row * D;
      float s = 0.f;
      for (int i = 0; i < D; ++i) s += xr[i];
      const float mean = s / (float)D;
      float v = 0.f;
      for (int i = 0; i < D; ++i) { float d0 = xr[i] - mean; v += d0 * d0; }
      const float rstd = rsqrtf(v / (float)D + 1e-5f);
      for (int i = 0; i < D; ++i) xr[i] = (xr[i] - mean) * rstd * gg[i] + bg[i];
    }
  }
  __syncthreads();
}

// One directional Mamba: x(LDS, L x D) -> mo += out_proj(y) [rows rev if rev]
template <int L, int D, int DTR>
__device__ void mamba_block(lfp x, lfp xz, lfp xi, lfp xdbl,
                            lfp dt, lfp mo,
                            const float* inp,   const float* convw,
                            const float* convb, const float* xproj,
                            const float* dtw,   const float* dtb,
                            const float* Alog,  const float* Dp,
                            const float* outp,  bool rev)
{
  const int tid = threadIdx.x;
  constexpr int W2 = 2 * D;
  constexpr int XW = DTR + 2 * D_STATE;

  // xz = x @ in_proj^T
  gemm_lds<0, L, D, W2>(x, D, inp, nullptr, xz, W2);
  __syncthreads();

  // depthwise causal conv (D_CONV=2) + SiLU -> xi
  {
    gfp cw = as_global(convw), cb = as_global(convb);
    for (int i = tid; i < L * D; i += NTHREADS) {
      const int l = i / D, d = i % D;
      const float cur = xz[l * W2 + d];
      const float prv = (l > 0) ? xz[(l - 1) * W2 + d] : 0.f;
      const float xc = prv * cw[d * 2 + 0] + cur * cw[d * 2 + 1] + cb[d];
      xi[i] = siluf(xc);
    }
  }
  __syncthreads();

  // xdbl = xi @ x_proj^T   (dt_in | B | C)
  gemm_lds<0, L, D, XW>(xi, D, xproj, nullptr, xdbl, XW);
  __syncthreads();

  // dt = softplus(xdbl[:, :DTR] @ dt_w^T + dt_b)
  gemm_lds<3, L, DTR, D>(xdbl, XW, dtw, dtb, dt, D);
  __syncthreads();

  // selective scan, one thread per channel; overwrite xi with gated y
  if (tid < D) {
    const int d = tid;
    gfp Ag = as_global(Alog);
    float A[D_STATE], h[D_STATE];
#pragma unroll
    for (int n = 0; n < D_STATE; ++n) {
      A[n] = -__expf(Ag[d * D_STATE + n]);
      h[n] = 0.f;
    }
    const float Dd = as_global(Dp)[d];
    for (int l = 0; l < L; ++l) {
      const float dtv = dt[l * D + d];
      const float xv  = xi[l * D + d];
      const float zv  = xz[l * W2 + D + d];
      lfp Bm = &xdbl[l * XW + DTR];
      lfp Cm = Bm + D_STATE;
      const float du = dtv * xv;
      float y = 0.f;
#pragma unroll
      for (int n = 0; n < D_STATE; ++n) {
        h[n] = __expf(dtv * A[n]) * h[n] + du * Bm[n];
        y += h[n] * Cm[n];
      }
      y = (y + xv * Dd) * siluf(zv);
      xi[l * D + d] = y;
    }
  }
  __syncthreads();

  // mo (+)= y @ out_proj^T (un-flip rows for the backward direction)
  gemm_acc_rev<L, D, D>(xi, D, outp, mo, D, rev);
  __syncthreads();
}

// FFN fused per 16-row tile through an LDS hidden tile, residual into x.
template <int L, int D>
__device__ void ffn_block(lfp x, lfp hbuf,
                          const float* c1w, const float* c1b,
                          const float* c2w, const float* c2b)
{
  const int wid  = threadIdx.x >> 5;
  const int lane = threadIdx.x & 31;
  const int r = lane & 15, hh = lane >> 4;
  constexpr int mt = (L + 15) >> 4;
  gfp c1bg = as_global(c1b), c2bg = as_global(c2b);
  for (int im = 0; im < mt; ++im) {
    // H = gelu(Xtile @ c1w^T + c1b) -> hbuf[16 x D_FF]
    for (int t = wid; t < D_FF / 16; t += NWAVE) {
      v8f acc = {};
      acc = wmma16<D>(x + im * 16 * D, D, c1w + t * 16 * D, D, 16, acc);
      const int n = t * 16 + r;
      const float bv = c1bg[n];
#pragma unroll
      for (int v = 0; v < 8; ++v)
        hbuf[(v + 8 * hh) * D_FF + n] = geluf(acc[v] + bv);
    }
    __syncthreads();
    // x_tile += H @ c2w^T + c2b
    constexpr int nt = (D + 15) >> 4;
    for (int t = wid; t < nt; t += NWAVE) {
      v8f acc = {};
      acc = wmma16<D_FF>(hbuf, D_FF, c2w + t * 16 * D_FF, D_FF,
                         D - t * 16, acc);
      const int n = t * 16 + r;
      if (n < D) {
        const float bv = c2bg[n];
#pragma unroll
        for (int v = 0; v < 8; ++v) {
          const int m = im * 16 + v + 8 * hh;
          if (m < L) x[m * D + n] += acc[v] + bv;
        }
      }
    }
    __syncthreads();
  }
}

template <int L, int D, int DTR>
__device__ void enc_layer(lfp x, lfp xf, lfp xz, lfp xi,
                          lfp xdbl, lfp dt, lfp mo, lfp hbuf,
                          const LayerW& w)
{
  const int tid = threadIdx.x;
  for (int i = tid; i < L * D; i += NTHREADS) mo[i] = 0.f;
  for (int i = tid; i < L * D; i += NTHREADS) {
    const int l = i / D, d = i % D;
    xf[(L - 1 - l) * D + d] = x[i];
  }
  __syncthreads();

  mamba_block<L, D, DTR>(x,  xz, xi, xdbl, dt, mo,
                         w.inp_f, w.convw_f, w.convb_f, w.xproj_f,
                         w.dtw_f, w.dtb_f, w.Alog_f, w.D_f, w.outp_f, false);
  mamba_block<L, D, DTR>(xf, xz, xi, xdbl, dt, mo,
                         w.inp_b, w.convw_b, w.convb_b, w.xproj_b,
                         w.dtw_b, w.dtb_b, w.Alog_b, w.D_b, w.outp_b, true);

  for (int i = tid; i < L * D; i += NTHREADS) x[i] += mo[i];
  __syncthreads();
  layernorm_rows<L, D>(x, w.ln1g, w.ln1b);
  ffn_block<L, D>(x, hbuf, w.c1w, w.c1b, w.c2w, w.c2b);
  layernorm_rows<L, D>(x, w.ln2g, w.ln2b);
}

// ---------------------------------------------------------------------------

__global__ __launch_bounds__(NTHREADS)
void smamba_forward(ModelP P)
{
  extern __shared__ float ldsraw[];
  lfp base = as_lds(ldsraw);
  lfp x    = base + OFF_X;
  lfp xz   = base + OFF_XZ;
  lfp xi   = base + OFF_XI;
  lfp dt   = base + OFF_DT;
  lfp mo   = base + OFF_MO;
  lfp xf   = base + OFF_XF;
  lfp xdbl = base + OFF_XDBL;
  lfp hbuf = base + OFF_H;
  lfp red  = base + OFF_RED;

  const int b   = blockIdx.x;
  const int tid = threadIdx.x;
  const int wid = tid >> 5, lane = tid & 31;
  const int r = lane & 15, hh = lane >> 4;

  // ---- patch embedding: x[c, d] = x_enc[b, :, c] @ emb_w[d, :] + emb_b + pos
  {
    gfp xe  = as_global(P.x_enc) + (size_t)b * SEQ_LEN * ENC_IN;
    gfp ew  = as_global(P.emb_w);
    gfp ebg = as_global(P.emb_b);
    gfp pg  = as_global(P.pos);
    const int rc = (r < ENC_IN) ? r : 0;   // clamp channel row
    for (int t = wid; t < D_MODEL / 16; t += NWAVE) {
      v8f acc = {};
      gfp Wrow = ew + (t * 16 + r) * SEQ_LEN;
#pragma unroll
      for (int k0 = 0; k0 < SEQ_LEN; k0 += 4) {
        const int ka = k0 + 2 * hh;
        v2f a = { xe[ka * ENC_IN + rc], xe[(ka + 1) * ENC_IN + rc] };
        v2f bb = *(gv2fp)(Wrow + ka);
        acc = __builtin_amdgcn_wmma_f32_16x16x4_f32(false, a, false, bb,
                                                    (short)0, acc, false, false);
      }
      const int n = t * 16 + r;
      const float add = ebg[n] + pg[n];
#pragma unroll
      for (int v = 0; v < 8; ++v) {
        const int c = v + 8 * hh;
        if (c < ENC_IN) x[c * D_MODEL + n] = acc[v] + add;
      }
    }
  }
  __syncthreads();

  // ---- time encoder (L=14 tokens of d=256)
  for (int li = 0; li < 2; ++li)
    enc_layer<ENC_IN, D_MODEL, DT_T>(x, xf, xz, xi, xdbl, dt, mo, hbuf, P.tl[li]);
  layernorm_rows<ENC_IN, D_MODEL>(x, P.time_ng, P.time_nb);

  // ---- transpose (14 x 256) -> (256 x 14), mo as temp
  for (int i = tid; i < ENC_IN * D_MODEL; i += NTHREADS) {
    const int c = i / D_MODEL, m = i % D_MODEL;
    mo[m * ENC_IN + c] = x[i];
  }
  __syncthreads();
  for (int i = tid; i < ENC_IN * D_MODEL; i += NTHREADS) x[i] = mo[i];
  __syncthreads();

  // ---- space encoder (L=256 tokens of d=14)
  for (int li = 0; li < 2; ++li)
    enc_layer<D_MODEL, SP_D, DT_S>(x, xf, xz, xi, xdbl, dt, mo, hbuf, P.sl[li]);
  layernorm_rows<D_MODEL, SP_D>(x, P.space_ng, P.space_nb);

  // ---- final projection: out[b] = flat . fin_w + fin_b
  {
    gfp fw = as_global(P.fin_w);
    float part = 0.f;
    for (int i = tid; i < ENC_IN * D_MODEL; i += NTHREADS)
      part += x[i] * fw[i];
#pragma unroll
    for (int o = 16; o > 0; o >>= 1) part += __shfl_xor(part, o, 32);
    if (lane == 0) red[wid] = part;
    __syncthreads();
    if (tid == 0) {
      float s = 0.f;
#pragma unroll
      for (int wv = 0; wv < NWAVE; ++wv) s += red[wv];
      P.out[b] = s + as_global(P.fin_b)[0];
    }
  }
}

// ---------------------------------------------------------------------------
// Host side: inputs arrive in JAX pytree (sorted-key) flatten order.
// ---------------------------------------------------------------------------

static void fill_layer(LayerW& w, void* const* d, int i)
{
  const float** q = (const float**)&w;
  for (int k = 0; k < 26; ++k) q[k] = (const float*)d[i + k];
  // order: c1b c1w c2b c2w ln1b ln1g ln2b ln2g |
  //        mb: A_log D conv_b conv_w dt_b dt_w in_proj out_proj x_proj |
  //        mf: A_log D conv_b conv_w dt_b dt_w in_proj out_proj x_proj
}

extern "C" void kernel_launch(void* const* d_in, const int* in_sizes, int n_in,
                              void* d_out, int out_size, void* d_ws, size_t ws_size,
                              hipStream_t stream)
{
  (void)in_sizes; (void)n_in; (void)out_size; (void)d_ws; (void)ws_size;

  ModelP P;
  P.emb_b = (const float*)d_in[0];
  P.emb_w = (const float*)d_in[1];
  P.fin_b = (const float*)d_in[2];
  P.fin_w = (const float*)d_in[3];
  P.pos   = (const float*)d_in[4];
  fill_layer(P.sl[0], d_in, 5);
  fill_layer(P.sl[1], d_in, 31);
  P.space_nb = (const float*)d_in[57];
  P.space_ng = (const float*)d_in[58];
  fill_layer(P.tl[0], d_in, 59);
  fill_layer(P.tl[1], d_in, 85);
  P.time_nb = (const float*)d_in[111];
  P.time_ng = (const float*)d_in[112];
  P.x_enc   = (const float*)d_in[113];
  P.out     = (float*)d_out;

  const size_t shmem = (size_t)LDS_FLOATS * sizeof(float); // 134 KB dynamic LDS
  smamba_forward<<<B_SZ, NTHREADS, shmem, stream>>>(P);
}